// AllPassMORRCirculantConv2d_360777252972
// MI455X (gfx1250) — compile-verified
//
#include <hip/hip_runtime.h>

typedef __attribute__((ext_vector_type(2))) float v2f;
typedef __attribute__((ext_vector_type(8))) float v8f;

namespace {
constexpr float kA = 0.8578f;
constexpr float kR = 0.8985f;
constexpr int IN_C = 32, OUT_C = 64, MB = 8;
constexpr int Q = 36, P = 8;
constexpr int H = 64, W = 64;
}

__global__ __launch_bounds__(256)
void morr_conv2d_kernel(const float* __restrict__ x,
                        const float* __restrict__ wgt,
                        const float* __restrict__ mos,
                        float* __restrict__ out)
{
    // 33 KB input halo tile [c(32)][r(4)][col(66)]; reused as sout[64][129] at the end
    __shared__ float s_tile[32 * 4 * 66];          // 8448 floats
    __shared__ float s_w[P * Q * MB];              // 2304 floats, weight * (gain/2pi)
    __shared__ float s_scale[Q];                   // 36 signed scales * Kc
    __shared__ int   s_abase[288];                 // im2col LDS base addr per win index

    const int tid = threadIdx.x;
    const int blk = blockIdx.x;                    // 256 blocks = 8 batches * 32 row-pairs
    const int b   = blk >> 5;
    const int oh0 = (blk & 31) * 2;

    // Fold MORR_GAIN and the cos() revolutions conversion into the weights:
    // phase_rev = sum_j (x^2) * (w * gain/2pi)  ->  v_cos_f32 input directly.
    const float wfold = (10.0f / 6.0f) * 0.15915493667125702f;   // gain * 1/(2pi)
    // Fold Kc into the signed scale vector: out = sum_q scale_q*Kc * rcp(den_q)
    const float Kc = -((1.0f - kA * kA) * (1.0f - kR * kR));

    // ---- stage prescaled weight (2304 = 9*256) ----
    #pragma unroll
    for (int i = 0; i < 9; ++i) s_w[tid + i * 256] = wgt[tid + i * 256] * wfold;

    // ---- signed scale vector (Q even: [half, -half]) with Kc folded in ----
    if (tid < Q) s_scale[tid] = ((tid < Q / 2) ? mos[tid] : -mos[tid - Q / 2]) * Kc;

    // ---- im2col address table: win -> LDS offset (lane-independent part) ----
    if (tid < 288) {
        int c  = tid / 9;
        int r9 = tid - c * 9;
        int ki = r9 / 3;
        int kj = r9 - ki * 3;
        s_abase[tid] = c * 264 + ki * 66 + kj;
    }

    // ---- stage zero-padded x tile (8448 = 33*256) ----
    const float* xb = x + (size_t)b * (IN_C * H * W);
    #pragma unroll
    for (int i = 0; i < 33; ++i) {
        int idx = tid + i * 256;
        int c   = idx / 264;
        int rem = idx - c * 264;
        int r   = rem / 66;
        int col = rem - r * 66;
        int ih  = oh0 - 1 + r;
        int iw  = col - 1;
        float v = 0.0f;
        if (ih >= 0 && ih < H && iw >= 0 && iw < W)
            v = xb[(c * H + ih) * W + iw];
        s_tile[idx] = v;
    }
    __syncthreads();

    const int lane = tid & 31;
    const int wv   = tid >> 5;          // wave 0..7
    const int hi   = lane >> 4;         // lane half (selects K pairs in A/B frags)
    const int lo   = lane & 15;
    const int wrow = wv >> 2;           // which of the 2 output rows
    const int owb  = (wv & 3) * 16;     // 16-column strip

    const int laneoff = wrow * 66 + owb + lo;
    // B-fragment per-lane constants: n = 16t+lo -> p = 2t + (lo>>3), k = lo&7
    const int kmod = lo & 7;
    const int phh  = lo >> 3;
    // j for fragment element (ks,v): j = 4*ks + 2*hi + v  (loop-invariant)
    const int j00 = 2 * hi,     j01 = 2 * hi + 1;
    const int j10 = 4 + 2 * hi, j11 = 5 + 2 * hi;
    const int kk00 = (kmod - j00) & 7, kk01 = (kmod - j01) & 7;
    const int kk10 = (kmod - j10) & 7, kk11 = (kmod - j11) & 7;

    const float twoAR = 2.0f * kA * kR;
    const float Dc    = 1.0f + (kA * kR) * (kA * kR);

    v8f acc0 = {}, acc1 = {}, acc2 = {}, acc3 = {};

    for (int q = 0; q < Q; ++q) {
        const float sq = s_scale[q];
        const int   wq = q * 8;

        // A fragments: plain x^2 (gain folded into weights), 16x4 fp32 layout
        float xa00 = s_tile[s_abase[wq + j00] + laneoff];
        float xa01 = s_tile[s_abase[wq + j01] + laneoff];
        float xa10 = s_tile[s_abase[wq + j10] + laneoff];
        float xa11 = s_tile[s_abase[wq + j11] + laneoff];
        v2f a0, a1;
        a0[0] = xa00 * xa00; a0[1] = xa01 * xa01;
        a1[0] = xa10 * xa10; a1[1] = xa11 * xa11;

        #pragma unroll
        for (int t = 0; t < 4; ++t) {
            // circulant B fragment built from prescaled weight: W'[p][q][(k-j)&7]
            const int base = ((2 * t + phh) * Q + q) * 8;
            v2f b0, b1;
            b0[0] = s_w[base + kk00]; b0[1] = s_w[base + kk01];
            b1[0] = s_w[base + kk10]; b1[1] = s_w[base + kk11];

            v8f phs = {};
            phs = __builtin_amdgcn_wmma_f32_16x16x4_f32(
                false, a0, false, b0, (short)0, phs, false, false);
            phs = __builtin_amdgcn_wmma_f32_16x16x4_f32(
                false, a1, false, b1, (short)0, phs, false, false);

            v8f* accp = (t == 0) ? &acc0 : (t == 1) ? &acc1 : (t == 2) ? &acc2 : &acc3;
            #pragma unroll
            for (int e = 0; e < 8; ++e) {
                float cv  = __builtin_amdgcn_cosf(phs[e]);   // phase already in revolutions
                float den = fmaf(-twoAR, cv, Dc);            // 1+(AR)^2 - 2AR cos(phi)
                (*accp)[e] += sq * __builtin_amdgcn_rcpf(den);
            }
        }
    }

    __syncthreads();  // input tile no longer needed; reuse LDS for output transpose

    #define STORE_TILE(T, ACC)                                           \
        { int ch = 16 * (T) + lo;                                        \
          _Pragma("unroll")                                              \
          for (int r = 0; r < 8; ++r) {                                  \
              int pl = wrow * 64 + owb + r + 8 * hi;                     \
              s_tile[ch * 129 + pl] = (ACC)[r];                          \
          } }
    STORE_TILE(0, acc0)
    STORE_TILE(1, acc1)
    STORE_TILE(2, acc2)
    STORE_TILE(3, acc3)
    #undef STORE_TILE
    __syncthreads();

    // coalesced store: out[b][ch][oh0+row][ow], 128 contiguous floats per channel
    float* ob = out + (size_t)b * (OUT_C * H * W) + oh0 * W;
    #pragma unroll
    for (int i = 0; i < 32; ++i) {
        int lin = tid + i * 256;
        int ch  = lin >> 7;
        int pl  = lin & 127;
        ob[ch * (H * W) + pl] = s_tile[ch * 129 + pl];
    }
}

extern "C" void kernel_launch(void* const* d_in, const int* in_sizes, int n_in,
                              void* d_out, int out_size, void* d_ws, size_t ws_size,
                              hipStream_t stream) {
    (void)in_sizes; (void)n_in; (void)d_ws; (void)ws_size; (void)out_size;
    const float* x   = (const float*)d_in[0];   // (8,32,64,64) f32
    const float* wgt = (const float*)d_in[1];   // (8,36,8) f32
    const float* mos = (const float*)d_in[2];   // (19,) f32
    float* out = (float*)d_out;                 // (8,64,64,64) f32

    dim3 grid(256), block(256);
    hipLaunchKernelGGL(morr_conv2d_kernel, grid, block, 0, stream, x, wgt, mos, out);
}